// ParametricScoreMLP_88759794139160
// MI455X (gfx1250) — compile-verified
//
#include <hip/hip_runtime.h>
#include <hip/hip_bf16.h>

// ParametricScoreMLP for MI455X (gfx1250, wave32).
//  Phase 2 (listed first for disasm visibility): pairwise relu-reduction on
//    VALU; c handled in pairs with packed fp32 VOP3P ops forced via inline asm
//    (v_pk_add_f32 + tied-accumulator v_pk_fma_f32) since the relu inside the
//    c-reduction cannot be expressed as a matmul.
//  Phase 1: q/k projections on f32 WMMA (V_WMMA_F32_16X16X4_F32) -> d_ws.
// Shapes: B=2 H=4 T=512 D=64 HID=128. ws usage: 2 * 4096*128 * 4B = 4 MB.

typedef __attribute__((ext_vector_type(2))) float v2f;
typedef __attribute__((ext_vector_type(8))) float v8f;

#define NROWS 4096   // B*H*T
#define DQK   64
#define HID   128

// Packed fp32 VOP3P helpers (CDNA5 ISA §15.10 opcodes 41 / 31).
// "v" on a 64-bit value allocates an even-aligned VGPR pair.
static __device__ __forceinline__ v2f pk_add(v2f a, v2f b) {
    v2f d;
    asm("v_pk_add_f32 %0, %1, %2" : "=v"(d) : "v"(a), "v"(b));
    return d;
}
// Tied accumulator: dst == src2, so the packed FMA accumulates in place and
// the register allocator inserts no v_mov copies around the loop-carried acc.
static __device__ __forceinline__ void pk_fma_acc(v2f& acc, v2f a, v2f b) {
    asm("v_pk_fma_f32 %0, %1, %2, %0" : "+v"(acc) : "v"(a), "v"(b));
}

// ---------------------------------------------------------------------------
// Phase 2: s[bh,i,j] = sum_c w2[c]*relu(qh[i,c]+kh[j,c]) + b2.
// 512 blocks: blockIdx -> (bh, 64x64 tile of the 512x512 score matrix).
// LDS row stride 130 words (pad 2) -> conflict-free b64 accesses; 4x4
// register blocking per thread; c in pairs: 4 VALU issues per 2 elements
// (v_pk_add_f32 + 2x v_max_f32 + tied v_pk_fma_f32).
// ---------------------------------------------------------------------------
#define LSTR 130

__global__ __launch_bounds__(256) void psm_pairwise(
    const float* __restrict__ qh, const float* __restrict__ kh,
    const float* __restrict__ W2, const float* __restrict__ b2,
    float* __restrict__ out)
{
    __shared__ __align__(16) float qs[64 * LSTR];
    __shared__ __align__(16) float ks[64 * LSTR];
    __shared__ __align__(16) float w2s[HID];

    const int t    = threadIdx.x;
    const int bh   = blockIdx.x >> 6;        // 0..7
    const int tile = blockIdx.x & 63;        // 8x8 tiles of 64x64
    const int ti   = (tile >> 3) * 64;
    const int tj   = (tile & 7) * 64;

    const float* qbase = qh + ((size_t)bh * 512 + ti) * HID;
    const float* kbase = kh + ((size_t)bh * 512 + tj) * HID;

    // Stage 64x128 slabs of qh and kh into LDS (coalesced 16B global loads,
    // 8B LDS stores because row stride 130 is only 8B aligned for odd rows).
#pragma unroll
    for (int iter = 0; iter < 8; ++iter) {
        const int idx = iter * 256 + t;
        const int i   = idx >> 5;            // 0..63
        const int c   = (idx & 31) * 4;      // 0..124
        const float4 gq = *(const float4*)(qbase + (size_t)i * HID + c);
        const float4 gk = *(const float4*)(kbase + (size_t)i * HID + c);
        *(v2f*)(qs + i * LSTR + c)     = (v2f){gq.x, gq.y};
        *(v2f*)(qs + i * LSTR + c + 2) = (v2f){gq.z, gq.w};
        *(v2f*)(ks + i * LSTR + c)     = (v2f){gk.x, gk.y};
        *(v2f*)(ks + i * LSTR + c + 2) = (v2f){gk.z, gk.w};
    }
    if (t < 32) *(float4*)(w2s + t * 4) = *(const float4*)(W2 + t * 4);
    __syncthreads();

    const int i0 = (t >> 4) * 4;
    const int j0 = (t & 15) * 4;

    v2f acc2[4][4];
#pragma unroll
    for (int ii = 0; ii < 4; ++ii)
#pragma unroll
        for (int jj = 0; jj < 4; ++jj) acc2[ii][jj] = (v2f){0.f, 0.f};

    for (int c0 = 0; c0 < HID; c0 += 4) {
        // two c-pairs per chunk: {c0,c0+1} and {c0+2,c0+3}
        v2f wp[2];
        wp[0] = *(const v2f*)(w2s + c0);
        wp[1] = *(const v2f*)(w2s + c0 + 2);
        v2f qa[4][2], kb[4][2];
#pragma unroll
        for (int r = 0; r < 4; ++r) {
            qa[r][0] = *(const v2f*)(qs + (i0 + r) * LSTR + c0);
            qa[r][1] = *(const v2f*)(qs + (i0 + r) * LSTR + c0 + 2);
            kb[r][0] = *(const v2f*)(ks + (j0 + r) * LSTR + c0);
            kb[r][1] = *(const v2f*)(ks + (j0 + r) * LSTR + c0 + 2);
        }
#pragma unroll
        for (int p = 0; p < 2; ++p) {
            const v2f w = wp[p];
#pragma unroll
            for (int ii = 0; ii < 4; ++ii) {
#pragma unroll
                for (int jj = 0; jj < 4; ++jj) {
                    v2f s = pk_add(qa[ii][p], kb[jj][p]);   // v_pk_add_f32
                    s.x = fmaxf(s.x, 0.0f);                 // v_max_f32 (VOPD-pairable)
                    s.y = fmaxf(s.y, 0.0f);
                    pk_fma_acc(acc2[ii][jj], s, w);         // v_pk_fma_f32 (tied)
                }
            }
        }
    }

    const float bias = b2[0];
#pragma unroll
    for (int r = 0; r < 4; ++r) {
        float4 o;
        o.x = acc2[r][0].x + acc2[r][0].y + bias;
        o.y = acc2[r][1].x + acc2[r][1].y + bias;
        o.z = acc2[r][2].x + acc2[r][2].y + bias;
        o.w = acc2[r][3].x + acc2[r][3].y + bias;
        *(float4*)(out + ((size_t)bh * 512 + ti + i0 + r) * 512 + tj + j0) = o;
    }
}

// ---------------------------------------------------------------------------
// Phase 1: one wave computes a 16-row x 128-col strip of qh (side 0) or kh
// (side 1) using 16x16x4 f32 WMMA, accumulating over D=64 in K-chunks of 4.
// kh gets b1 folded in.  512 waves total = 64 blocks * 8 waves.
// ---------------------------------------------------------------------------
__global__ __launch_bounds__(256) void psm_proj_wmma(
    const float* __restrict__ q, const float* __restrict__ k,
    const float* __restrict__ W1, const float* __restrict__ b1,
    float* __restrict__ qh, float* __restrict__ kh)
{
    const int wave = blockIdx.x * 8 + (threadIdx.x >> 5);
    const int lane = threadIdx.x & 31;
    const int side = wave >> 8;            // 0 -> q, 1 -> k
    const int rt   = wave & 255;           // 16-row tile index within 4096 rows
    const int rowbase = rt * 16;
    const int m     = lane & 15;
    const int khalf = lane >> 4;           // A/B K sub-pair select (ISA layout)

    const float* src   = side ? k : q;
    const float* Wb    = W1 + side * DQK * HID;  // W1 is [2D=128][HID=128] row-major
    float*       dst   = side ? kh : qh;

    v8f acc[8];
#pragma unroll
    for (int nt = 0; nt < 8; ++nt) acc[nt] = (v8f){0,0,0,0,0,0,0,0};

    const float* arow = src + (size_t)(rowbase + m) * DQK;

#pragma unroll 4
    for (int kc = 0; kc < 16; ++kc) {
        const int k0 = kc * 4 + khalf * 2;
        // A 16x4 f32: lanes0-15 hold K={4kc,4kc+1}, lanes16-31 hold K={4kc+2,4kc+3}
        v2f a;
        a.x = arow[k0];
        a.y = arow[k0 + 1];
        const float* w0 = Wb + (size_t)k0 * HID;
#pragma unroll
        for (int nt = 0; nt < 8; ++nt) {
            const int n = nt * 16 + m;
            // B 4x16 f32: mirrored K striping, N = lane&15
            v2f b;
            b.x = w0[n];
            b.y = w0[HID + n];
            acc[nt] = __builtin_amdgcn_wmma_f32_16x16x4_f32(
                /*neg_a=*/false, a, /*neg_b=*/false, b,
                /*c_mod=*/(short)0, acc[nt],
                /*reuse_a=*/false, /*reuse_b=*/false);
        }
    }

    // C/D layout: VGPR r <-> M = r (lanes0-15) or M = r+8 (lanes16-31), N = lane&15
#pragma unroll
    for (int nt = 0; nt < 8; ++nt) {
        const int c = nt * 16 + m;
        const float bias = side ? b1[c] : 0.0f;
#pragma unroll
        for (int r = 0; r < 8; ++r) {
            const int row = rowbase + khalf * 8 + r;
            dst[(size_t)row * HID + c] = acc[nt][r] + bias;
        }
    }
}

// ---------------------------------------------------------------------------
extern "C" void kernel_launch(void* const* d_in, const int* in_sizes, int n_in,
                              void* d_out, int out_size, void* d_ws, size_t ws_size,
                              hipStream_t stream) {
    const float* q  = (const float*)d_in[0];  // [2,4,512,64]
    const float* k  = (const float*)d_in[1];  // [2,4,512,64]
    const float* W1 = (const float*)d_in[2];  // [128,128]
    const float* b1 = (const float*)d_in[3];  // [128]
    const float* W2 = (const float*)d_in[4];  // [128,1]
    const float* b2 = (const float*)d_in[5];  // [1]
    float* out = (float*)d_out;               // [2,4,512,512]

    float* qh = (float*)d_ws;                 // 4096*128 f32 = 2 MB
    float* kh = qh + (size_t)NROWS * HID;     // + 2 MB (ws must be >= 4 MB)

    psm_proj_wmma<<<64, 256, 0, stream>>>(q, k, W1, b1, qh, kh);
    psm_pairwise<<<512, 256, 0, stream>>>(qh, kh, W2, b2, out);
}